// SinkhornLoss_29137058136258
// MI455X (gfx1250) — compile-verified
//
#include <hip/hip_runtime.h>

// CDNA5 / gfx1250, wave32. One V_WMMA_F32_16X16X4_F32 per 16x16 cost tile.
typedef __attribute__((ext_vector_type(2))) float v2f;
typedef __attribute__((ext_vector_type(8))) float v8f;

#define NPTS    6400
#define NT      400          // number of 16-wide tiles
#define EPSV    0.0025f      // blur^2 = 0.05^2
#define INV_EPS 400.0f
#define LOGW    (-8.7640533f) // -log(6400)
#define NEG_BIG (-1e30f)

__device__ __forceinline__ v8f wmma_tile(v2f a, v2f b) {
  v8f c = {0.f,0.f,0.f,0.f,0.f,0.f,0.f,0.f};
  // 8-arg form: (neg_a, A, neg_b, B, c_mod, C, reuse_a, reuse_b)
  return __builtin_amdgcn_wmma_f32_16x16x4_f32(false, a, false, b, (short)0, c,
                                               false, false);
}

// Build packed point records: (c0, c1, c2, q) with q = 0.5*|p|^2 / eps.
__global__ void k_setup(const float* __restrict__ sk, const float* __restrict__ tg,
                        const float* __restrict__ pos,
                        float4* __restrict__ pX, float4* __restrict__ pY) {
  int i = blockIdx.x * blockDim.x + threadIdx.x;
  if (i >= NPTS) return;
  float gy = pos[2*i], gx = pos[2*i+1];
  float v = sk[i];
  pX[i] = make_float4(v, gy, gx, 0.5f*(v*v + gy*gy + gx*gx)*INV_EPS);
  float w = tg[i];
  pY[i] = make_float4(w, gy, gx, 0.5f*(w*w + gy*gy + gx*gx)*INV_EPS);
}

__global__ void k_zero3(float* a, float* b, float* c) {
  int i = blockIdx.x * blockDim.x + threadIdx.x;
  if (i >= NPTS) return;
  a[i] = 0.f; b[i] = 0.f; c[i] = 0.f;
}

// out_i = -eps * logsumexp_j((pot_j + eps*logw - C_ij)/eps)
// C_ij recomputed via WMMA: score = x.y/eps + c_j, then +(-q_i) folded into lse.
// symMode: out_i = 0.5*(potPrev_i + softmin_i).
__global__ __launch_bounds__(256) void k_softmin(
    const float4* __restrict__ rowPts, const float4* __restrict__ colPts,
    const float* __restrict__ potCol, const float* __restrict__ potPrev,
    float* __restrict__ outPot, int symMode)
{
  __shared__ float lm[8][8][32];
  __shared__ float ls[8][8][32];
  const int wave = threadIdx.x >> 5;
  const int lane = threadIdx.x & 31;
  const int m    = lane & 15;
  const int hi   = lane >> 4;
  const int rbase = blockIdx.x * 16;

  // A fragment (16x4): rows = x points, K = (c0,c1,c2,1)
  float4 rv = rowPts[rbase + m];
  v2f a; a.x = hi ? rv.z : rv.x; a.y = hi ? 1.0f : rv.y;

  float mr[8], sr[8];
#pragma unroll
  for (int k = 0; k < 8; k++) { mr[k] = NEG_BIG; sr[k] = 0.f; }

  for (int ct = wave; ct < NT; ct += 8) {
    int cidx = ct * 16 + m;
    float4 cv = colPts[cidx];
    float cj = potCol[cidx] * INV_EPS + LOGW - cv.w;  // (pot_j)/eps + logw - q_j
    v2f b;
    b.x = (hi ? cv.z : cv.x) * INV_EPS;
    b.y = hi ? cj : cv.y * INV_EPS;                   // K=3 row carries bias c_j
    v8f d = wmma_tile(a, b);                          // d = x.y/eps + c_j
#pragma unroll
    for (int k = 0; k < 8; k++) {
      float v  = d[k];
      float mo = mr[k];
      float mn = fmaxf(mo, v);
      sr[k] = sr[k] * __expf(mo - mn) + __expf(v - mn);
      mr[k] = mn;
    }
  }
#pragma unroll
  for (int k = 0; k < 8; k++) { lm[wave][k][lane] = mr[k]; ls[wave][k][lane] = sr[k]; }
  __syncthreads();

  if (threadIdx.x < 16) {
    int r = threadIdx.x, k = r & 7, half = r >> 3;
    float M = NEG_BIG;
    for (int w = 0; w < 8; w++)
      for (int l = 0; l < 16; l++)
        M = fmaxf(M, lm[w][k][half*16 + l]);
    float S = 0.f;
    for (int w = 0; w < 8; w++)
      for (int l = 0; l < 16; l++)
        S += ls[w][k][half*16 + l] * __expf(lm[w][k][half*16 + l] - M);
    float q   = rowPts[rbase + r].w;                  // r_i = -q_i
    float val = -EPSV * (M + __logf(S) - q);
    if (symMode) val = 0.5f * (potPrev[rbase + r] + val);
    outPot[rbase + r] = val;
  }
}

// Accumulate, per row i:
//   oW = sum_j (e^{t+cA_j+rA_i} + e^{t+cB_j+rB_i}) * colPt_j   (3-vec)
//   oSB = sum_j e^{t+cB_j+rB_i}
// where t = x.y/eps (WMMA, bias slot zeroed), rX_i = potRX_i/eps + addRX - q_i,
// cX_j = potCX_j/eps + addCX - q_j. All exponents <= ~0 (normalized plans).
__global__ __launch_bounds__(256) void k_grad(
    const float4* __restrict__ rowPts, const float4* __restrict__ colPts,
    const float* __restrict__ potRA, float addRA,
    const float* __restrict__ potCA, float addCA,
    const float* __restrict__ potRB, float addRB,
    const float* __restrict__ potCB, float addCB,
    float* __restrict__ oW0, float* __restrict__ oW1, float* __restrict__ oW2,
    float* __restrict__ oSB)
{
  __shared__ float l0[8][8][32];
  __shared__ float l1[8][8][32];
  __shared__ float l2[8][8][32];
  __shared__ float l3[8][8][32];
  const int wave = threadIdx.x >> 5;
  const int lane = threadIdx.x & 31;
  const int m    = lane & 15;
  const int hi   = lane >> 4;
  const int rbase = blockIdx.x * 16;

  float rA8[8], rB8[8];
#pragma unroll
  for (int k = 0; k < 8; k++) {
    int rr = rbase + k + 8 * hi;                      // C/D row M = k + 8*hi
    float q = rowPts[rr].w;
    rA8[k] = potRA[rr] * INV_EPS + addRA - q;
    rB8[k] = potRB[rr] * INV_EPS + addRB - q;
  }
  float4 rv = rowPts[rbase + m];
  v2f a; a.x = hi ? rv.z : rv.x; a.y = hi ? 1.0f : rv.y;

  float w0[8], w1[8], w2[8], sB[8];
#pragma unroll
  for (int k = 0; k < 8; k++) { w0[k] = w1[k] = w2[k] = sB[k] = 0.f; }

  for (int ct = wave; ct < NT; ct += 8) {
    int cidx = ct * 16 + m;
    float4 cv = colPts[cidx];
    float cA = potCA[cidx] * INV_EPS + addCA - cv.w;
    float cB = potCB[cidx] * INV_EPS + addCB - cv.w;
    v2f b;
    b.x = (hi ? cv.z : cv.x) * INV_EPS;
    b.y = hi ? 0.0f : cv.y * INV_EPS;                 // bias slot = 0 -> d = t
    v8f d = wmma_tile(a, b);
#pragma unroll
    for (int k = 0; k < 8; k++) {
      float t  = d[k];
      float eA = __expf(t + cA + rA8[k]);
      float eB = __expf(t + cB + rB8[k]);
      float w  = eA + eB;
      w0[k] += w * cv.x; w1[k] += w * cv.y; w2[k] += w * cv.z; sB[k] += eB;
    }
  }
#pragma unroll
  for (int k = 0; k < 8; k++) {
    l0[wave][k][lane] = w0[k]; l1[wave][k][lane] = w1[k];
    l2[wave][k][lane] = w2[k]; l3[wave][k][lane] = sB[k];
  }
  __syncthreads();
  if (threadIdx.x < 16) {
    int r = threadIdx.x, k = r & 7, half = r >> 3;
    float s0 = 0, s1 = 0, s2 = 0, s3 = 0;
    for (int w = 0; w < 8; w++)
      for (int l = 0; l < 16; l++) {
        int ln = half*16 + l;
        s0 += l0[w][k][ln]; s1 += l1[w][k][ln];
        s2 += l2[w][k][ln]; s3 += l3[w][k][ln];
      }
    oW0[rbase + r] = s0; oW1[rbase + r] = s1;
    oW2[rbase + r] = s2; oSB[rbase + r] = s3;
  }
}

// grad_i = x_i*(sG_i - sX_i) - accY_i + accX_i ;  partial sums of |grad|^2
__global__ __launch_bounds__(256) void k_fin1(
    const float4* __restrict__ pX,
    const float* __restrict__ sG, const float* __restrict__ sX,
    const float* __restrict__ y0, const float* __restrict__ y1, const float* __restrict__ y2,
    const float* __restrict__ x0, const float* __restrict__ x1, const float* __restrict__ x2,
    float* __restrict__ partials)
{
  __shared__ float red[256];
  int i = blockIdx.x * 256 + threadIdx.x;
  float4 p = pX[i];
  float sc = sG[i] - sX[i];
  float g0 = p.x * sc - y0[i] + x0[i];
  float g1 = p.y * sc - y1[i] + x1[i];
  float g2 = p.z * sc - y2[i] + x2[i];
  red[threadIdx.x] = g0*g0 + g1*g1 + g2*g2;
  __syncthreads();
  for (int s = 128; s > 0; s >>= 1) {
    if (threadIdx.x < s) red[threadIdx.x] += red[threadIdx.x + s];
    __syncthreads();
  }
  if (threadIdx.x == 0) partials[blockIdx.x] = red[0];
}

__global__ void k_fin2(const float* __restrict__ partials, float* __restrict__ out) {
  if (threadIdx.x == 0) {
    float s = 0.f;
    for (int i = 0; i < 25; i++) s += partials[i];
    out[0] = s / 19200.0f;   // mean over B*H*W*3
  }
}

extern "C" void kernel_launch(void* const* d_in, const int* in_sizes, int n_in,
                              void* d_out, int out_size, void* d_ws, size_t ws_size,
                              hipStream_t stream) {
  (void)in_sizes; (void)n_in; (void)out_size; (void)ws_size;
  const float* sk  = (const float*)d_in[0];
  const float* tg  = (const float*)d_in[1];
  const float* pos = (const float*)d_in[2];
  float* ws = (float*)d_ws;

  float4* pX   = (float4*)(ws + 0);        // 6400*4
  float4* pY   = (float4*)(ws + 25600);    // 6400*4
  float* f     = ws + 51200;
  float* g     = ws + 57600;
  float* px0   = ws + 64000;
  float* px1   = ws + 70400;
  float* ffin  = ws + 76800;
  float* pxfin = ws + 83200;
  float* aY0   = ws + 89600;
  float* aY1   = ws + 96000;
  float* aY2   = ws + 102400;
  float* sG    = ws + 108800;
  float* aX0   = ws + 115200;
  float* aX1   = ws + 121600;
  float* aX2   = ws + 128000;
  float* sX    = ws + 134400;
  float* parts = ws + 140800;

  k_setup<<<25, 256, 0, stream>>>(sk, tg, pos, pX, pY);
  k_zero3<<<25, 256, 0, stream>>>(f, g, px0);

  // 20 Sinkhorn ab-steps: f = softmin(Cxy, g+elw); g = softmin(Cyx, f+elw)
  for (int it = 0; it < 20; ++it) {
    k_softmin<<<NT, 256, 0, stream>>>(pX, pY, g, nullptr, f, 0);
    k_softmin<<<NT, 256, 0, stream>>>(pY, pX, f, nullptr, g, 0);
  }
  // 20 symmetric steps on Cxx (ping-pong; ends in px0)
  float* pa = px0; float* pb = px1;
  for (int it = 0; it < 20; ++it) {
    k_softmin<<<NT, 256, 0, stream>>>(pX, pX, pa, pa, pb, 1);
    float* t = pa; pa = pb; pb = t;
  }
  // final differentiable softmins (g_fin == g already)
  k_softmin<<<NT, 256, 0, stream>>>(pX, pY, g, nullptr, ffin, 0);
  k_softmin<<<NT, 256, 0, stream>>>(pX, pX, pa, nullptr, pxfin, 0);

  // transport-plan accumulations: pi^f + (pi^g)^T over Cxy ; pi^x + (pi^x)^T over Cxx
  k_grad<<<NT, 256, 0, stream>>>(pX, pY,
                                 ffin, 0.f,  g, LOGW,    // A: pi^f_ij
                                 f, LOGW,    g, 0.f,     // B: pi^g_ji
                                 aY0, aY1, aY2, sG);
  k_grad<<<NT, 256, 0, stream>>>(pX, pX,
                                 pxfin, 0.f, pa, LOGW,   // A: pi^x_il
                                 pa, LOGW,   pxfin, 0.f, // B: pi^x_li
                                 aX0, aX1, aX2, sX);

  k_fin1<<<25, 256, 0, stream>>>(pX, sG, sX, aY0, aY1, aY2, aX0, aX1, aX2, parts);
  k_fin2<<<1, 32, 0, stream>>>(parts, (float*)d_out);
}